// GATv2_21045339750809
// MI455X (gfx1250) — compile-verified
//
#include <hip/hip_runtime.h>
#include <math.h>

#define N_NODES  50000
#define N_EDGES  800000
#define IN_FEATS 128
#define H_FEATS  64
#define HEADS    4
#define D_HEAD   16
#define NEG_SLOPE 0.2f

typedef __attribute__((ext_vector_type(2))) float v2f;
typedef __attribute__((ext_vector_type(8))) float v8f;

// ---------------------------------------------------------------------------
// CDNA5 fp32 WMMA: D(16x16,f32) = A(16x4,f32) * B(4x16,f32) + C
// 8-arg pattern per bridge doc: (neg_a, A, neg_b, B, c_mod, C, reuse_a, reuse_b)
// ---------------------------------------------------------------------------
__device__ __forceinline__ v8f wmma_f32_16x16x4(v2f a, v2f b, v8f c) {
    return __builtin_amdgcn_wmma_f32_16x16x4_f32(
        /*neg_a=*/false, a, /*neg_b=*/false, b,
        /*c_mod=*/(short)0, c, /*reuse_a=*/false, /*reuse_b=*/false);
}

// ---------------------------------------------------------------------------
// C[M,64] = A[M,K] @ W[K,64] + bias, with W supplied pre-transposed WT[64,K].
// Block = 128 threads = 4 waves; wave w owns the 16-column tile w, block owns
// one 16-row tile. M must be a multiple of 16 (50000 = 3125*16): EXEC all-1s.
// ISA 7.12.2 fp32 fragment layouts:
//   A 16x4 : lane holds row lane%16, K = (lane<16?0:2), (..+1)
//   B 4x16 : lane holds col lane%16, same K split (pre-transposed -> contig)
//   C 16x16: VGPR r -> row r + (lane<16?0:8), col lane%16
// ---------------------------------------------------------------------------
__global__ void gemm_bias_wmma(const float* __restrict__ A,
                               const float* __restrict__ WT,
                               const float* __restrict__ bias,
                               float* __restrict__ C, int K) {
    const int lane = threadIdx.x & 31;
    const int wave = threadIdx.x >> 5;        // 0..3 : 16-col tile
    const int mt   = blockIdx.x;              // 16-row tile
    const int rm   = lane & 15;
    const int koff = (lane >> 4) << 1;        // 0 or 2 within each K-group of 4

    const float* arow = A  + (size_t)(mt * 16 + rm) * K + koff;
    const float* brow = WT + (size_t)(wave * 16 + rm) * K + koff;

    v8f acc = {0.f, 0.f, 0.f, 0.f, 0.f, 0.f, 0.f, 0.f};
    for (int k = 0; k < K; k += 4) {
        const float2 av = *(const float2*)(arow + k);
        const float2 bv = *(const float2*)(brow + k);
        v2f a; a.x = av.x; a.y = av.y;
        v2f b; b.x = bv.x; b.y = bv.y;
        acc = wmma_f32_16x16x4(a, b, acc);
    }

    const int   col  = wave * 16 + rm;
    const float bv   = bias[col];
    const int   row0 = mt * 16 + ((lane >> 4) << 3);   // +8 for upper half-wave
#pragma unroll
    for (int r = 0; r < 8; ++r)
        C[(size_t)(row0 + r) * H_FEATS + col] = acc[r] + bv;
}

// ---------------------------------------------------------------------------
// Helpers
// ---------------------------------------------------------------------------
__global__ void transpose_k(const float* __restrict__ W, float* __restrict__ WT,
                            int R, int Ccols) {
    int t = blockIdx.x * blockDim.x + threadIdx.x;
    if (t >= R * Ccols) return;
    int r = t / Ccols, c = t - r * Ccols;
    WT[(size_t)c * R + r] = W[t];
}

__global__ void fill_k(float* __restrict__ p, float v, int n) {
    int t = blockIdx.x * blockDim.x + threadIdx.x;
    if (t < n) p[t] = v;
}

__global__ void relu_k(float* __restrict__ p, int n) {
    int t = blockIdx.x * blockDim.x + threadIdx.x;
    if (t < n) p[t] = fmaxf(p[t], 0.f);
}

// Bitwise float atomic-max (race-safe for any sign mix; init with -inf).
__device__ __forceinline__ void atomicMaxFloat(float* addr, float val) {
    if (val >= 0.f) atomicMax((int*)addr, __float_as_int(val));
    else            atomicMin((unsigned int*)addr, __float_as_uint(val));
}

// ---------------------------------------------------------------------------
// Edge phase: one thread per (edge, head)
// ---------------------------------------------------------------------------
__global__ void edge_score_max(const int* __restrict__ src,
                               const int* __restrict__ dst,
                               const float* __restrict__ fs,
                               const float* __restrict__ fd,
                               const float* __restrict__ attn_l,
                               float* __restrict__ score,
                               float* __restrict__ mmax) {
    int t = blockIdx.x * blockDim.x + threadIdx.x;
    if (t >= N_EDGES * HEADS) return;
    const int e = t >> 2, h = t & 3;
    const int s = src[e], d = dst[e];

    const float4* ps = (const float4*)(fs + (size_t)s * H_FEATS + h * D_HEAD);
    const float4* pd = (const float4*)(fd + (size_t)d * H_FEATS + h * D_HEAD);
    const float4* pa = (const float4*)(attn_l + h * D_HEAD);

    float acc = 0.f;
#pragma unroll
    for (int q = 0; q < 4; ++q) {
        float4 a = ps[q], b = pd[q], w = pa[q];
        float x0 = a.x + b.x, x1 = a.y + b.y, x2 = a.z + b.z, x3 = a.w + b.w;
        x0 = (x0 > 0.f) ? x0 : NEG_SLOPE * x0;
        x1 = (x1 > 0.f) ? x1 : NEG_SLOPE * x1;
        x2 = (x2 > 0.f) ? x2 : NEG_SLOPE * x2;
        x3 = (x3 > 0.f) ? x3 : NEG_SLOPE * x3;
        acc += x0 * w.x + x1 * w.y + x2 * w.z + x3 * w.w;
    }
    score[t] = acc;
    atomicMaxFloat(&mmax[(size_t)d * HEADS + h], acc);
}

__global__ void edge_exp_sum(const int* __restrict__ dst,
                             float* __restrict__ score,
                             const float* __restrict__ mmax,
                             float* __restrict__ ssum) {
    int t = blockIdx.x * blockDim.x + threadIdx.x;
    if (t >= N_EDGES * HEADS) return;
    const int e = t >> 2, h = t & 3;
    const int d = dst[e];
    float w = expf(score[t] - mmax[(size_t)d * HEADS + h]);
    score[t] = w;                                   // reuse buffer: exp values
    atomicAdd(&ssum[(size_t)d * HEADS + h], w);
}

__global__ void edge_aggregate(const int* __restrict__ src,
                               const int* __restrict__ dst,
                               const float* __restrict__ score,
                               const float* __restrict__ ssum,
                               const float* __restrict__ fs,
                               float* __restrict__ outacc) {
    int t = blockIdx.x * blockDim.x + threadIdx.x;
    if (t >= N_EDGES * HEADS) return;
    const int e = t >> 2, h = t & 3;
    const int s = src[e], d = dst[e];
    const float alpha = score[t] / ssum[(size_t)d * HEADS + h];
    const float* ps = fs + (size_t)s * H_FEATS + h * D_HEAD;
    float*       po = outacc + (size_t)d * H_FEATS + h * D_HEAD;
#pragma unroll
    for (int i = 0; i < D_HEAD; ++i)
        atomicAdd(&po[i], alpha * ps[i]);
}

// Final projection: logits[n,2] = h[n,:] @ W_out + b_out
__global__ void out_proj(const float* __restrict__ h,
                         const float* __restrict__ Wout,
                         const float* __restrict__ bout,
                         float* __restrict__ out) {
    int n = blockIdx.x * blockDim.x + threadIdx.x;
    if (n >= N_NODES) return;
    const float* ph = h + (size_t)n * H_FEATS;
    float a0 = bout[0], a1 = bout[1];
#pragma unroll
    for (int j = 0; j < H_FEATS; ++j) {
        float v = ph[j];
        a0 += v * Wout[j * 2 + 0];
        a1 += v * Wout[j * 2 + 1];
    }
    out[n * 2 + 0] = a0;
    out[n * 2 + 1] = a1;
}

// ---------------------------------------------------------------------------
extern "C" void kernel_launch(void* const* d_in, const int* in_sizes, int n_in,
                              void* d_out, int out_size, void* d_ws, size_t ws_size,
                              hipStream_t stream) {
    (void)in_sizes; (void)n_in; (void)out_size; (void)ws_size;

    const float* feature  = (const float*)d_in[0];
    const int*   src      = (const int*)  d_in[1];
    const int*   dst      = (const int*)  d_in[2];
    const float* W_in     = (const float*)d_in[3];
    const float* b_in     = (const float*)d_in[4];
    const float* fc_src_W = (const float*)d_in[5];
    const float* fc_src_b = (const float*)d_in[6];
    const float* fc_dst_W = (const float*)d_in[7];
    const float* fc_dst_b = (const float*)d_in[8];
    const float* attn     = (const float*)d_in[9];
    const float* W_out    = (const float*)d_in[10];
    const float* b_out    = (const float*)d_in[11];
    float* logits = (float*)d_out;

    // Workspace carve-up (floats)
    float* ws = (float*)d_ws;
    size_t off = 0;
    float* h     = ws + off; off += (size_t)N_NODES * H_FEATS;   // node feats / aggregation acc
    float* fs    = ws + off; off += (size_t)N_NODES * H_FEATS;
    float* fd    = ws + off; off += (size_t)N_NODES * H_FEATS;
    float* score = ws + off; off += (size_t)N_EDGES * HEADS;     // logits, then exp-values
    float* mmax  = ws + off; off += (size_t)N_NODES * HEADS;
    float* ssum  = ws + off; off += (size_t)N_NODES * HEADS;
    float* WT_in = ws + off; off += (size_t)H_FEATS * IN_FEATS;  // [64,128]
    float* WT_s  = ws + off; off += (size_t)2 * H_FEATS * H_FEATS;
    float* WT_d  = ws + off; off += (size_t)2 * H_FEATS * H_FEATS;

    const int mtiles = N_NODES / 16;             // 3125 exact
    const int EH     = N_EDGES * HEADS;          // 3.2M
    const int ETB    = 256;

    // Pre-transpose weights so WMMA B-fragments load contiguously.
    transpose_k<<<(IN_FEATS * H_FEATS + 255) / 256, 256, 0, stream>>>(W_in, WT_in, IN_FEATS, H_FEATS);
    for (int l = 0; l < 2; ++l) {
        transpose_k<<<(H_FEATS * H_FEATS + 255) / 256, 256, 0, stream>>>(
            fc_src_W + (size_t)l * H_FEATS * H_FEATS, WT_s + (size_t)l * H_FEATS * H_FEATS, H_FEATS, H_FEATS);
        transpose_k<<<(H_FEATS * H_FEATS + 255) / 256, 256, 0, stream>>>(
            fc_dst_W + (size_t)l * H_FEATS * H_FEATS, WT_d + (size_t)l * H_FEATS * H_FEATS, H_FEATS, H_FEATS);
    }

    // h = feature @ W_in + b_in   (K = 128)
    gemm_bias_wmma<<<mtiles, 128, 0, stream>>>(feature, WT_in, b_in, h, IN_FEATS);

    for (int l = 0; l < 2; ++l) {
        // fs/fd = h @ W + b  (K = 64)
        gemm_bias_wmma<<<mtiles, 128, 0, stream>>>(h, WT_s + (size_t)l * H_FEATS * H_FEATS,
                                                   fc_src_b + l * H_FEATS, fs, H_FEATS);
        gemm_bias_wmma<<<mtiles, 128, 0, stream>>>(h, WT_d + (size_t)l * H_FEATS * H_FEATS,
                                                   fc_dst_b + l * H_FEATS, fd, H_FEATS);

        // edge softmax state
        fill_k<<<(N_NODES * HEADS + 255) / 256, 256, 0, stream>>>(mmax, -INFINITY, N_NODES * HEADS);
        fill_k<<<(N_NODES * HEADS + 255) / 256, 256, 0, stream>>>(ssum, 0.f, N_NODES * HEADS);

        edge_score_max<<<(EH + ETB - 1) / ETB, ETB, 0, stream>>>(src, dst, fs, fd,
                                                                 attn + l * HEADS * D_HEAD, score, mmax);
        edge_exp_sum<<<(EH + ETB - 1) / ETB, ETB, 0, stream>>>(dst, score, mmax, ssum);

        // h is consumed by the two GEMMs above -> recycle as aggregation accumulator
        fill_k<<<(N_NODES * H_FEATS + 255) / 256, 256, 0, stream>>>(h, 0.f, N_NODES * H_FEATS);
        edge_aggregate<<<(EH + ETB - 1) / ETB, ETB, 0, stream>>>(src, dst, score, ssum, fs, h);
        relu_k<<<(N_NODES * H_FEATS + 255) / 256, 256, 0, stream>>>(h, N_NODES * H_FEATS);
    }

    out_proj<<<(N_NODES + 255) / 256, 256, 0, stream>>>(h, W_out, b_out, logits);
}